// NormBp_56607668961812
// MI455X (gfx1250) — compile-verified
//
#include <hip/hip_runtime.h>
#include <cstdint>
#include <cstddef>

// LIF neuron scan: y[t] = spike(v), v = 0.9*v + x[t], hard reset on spike.
// Memory-bound streaming scan (822 MB @ 23.3 TB/s ~= 35 us floor).
// CDNA5 path: async global->LDS staging (ASYNCcnt) double-buffered per wave,
// wave32 B128 coalescing, non-temporal stores for the write-once output.

typedef float v4f __attribute__((ext_vector_type(4)));

#define T_STEPS 16
#define BLOCK   256

__global__ __launch_bounds__(BLOCK) void lif_scan_async(
    const v4f* __restrict__ x, v4f* __restrict__ y, unsigned n4)
{
    // Double-buffered staging tile: 2 x 256 x 16B = 8 KB LDS per block.
    __shared__ v4f buf[2][BLOCK];

    const unsigned tid = threadIdx.x;
    const unsigned gid = blockIdx.x * BLOCK + tid;
    if (gid >= n4) return;

    // LDS byte address for this lane's slot in each buffer.
    // Flat pointers into the LDS aperture carry the LDS byte offset in the
    // low 32 bits, which is what the async
    // instruction's VDST operand expects.
    const unsigned lds0 = (unsigned)(uintptr_t)&buf[0][tid];
    const unsigned lds1 = (unsigned)(uintptr_t)&buf[1][tid];

    // Prologue: kick off async load of timestep 0 into buffer 0.
    {
        const v4f* g = x + gid;
        asm volatile("global_load_async_to_lds_b128 %0, %1, off"
                     :: "v"(lds0), "v"(g) : "memory");
    }

    v4f v = {0.0f, 0.0f, 0.0f, 0.0f};

#pragma unroll
    for (int t = 0; t < T_STEPS; ++t) {
        if (t + 1 < T_STEPS) {
            // Issue next timestep's async load into the other buffer, then
            // wait until only that one is outstanding -> buffer for t landed
            // (async loads complete in order).
            const v4f*     g   = x + (size_t)(t + 1) * n4 + gid;
            const unsigned lds = ((t + 1) & 1) ? lds1 : lds0;
            asm volatile("global_load_async_to_lds_b128 %0, %1, off"
                         :: "v"(lds), "v"(g) : "memory");
            asm volatile("s_wait_asynccnt 1" ::: "memory");
        } else {
            asm volatile("s_wait_asynccnt 0" ::: "memory");
        }

        // This wave only reads the LDS region it asynchronously filled
        // itself, so no workgroup barrier is required.
        v4f xv = buf[t & 1][tid];

        v4f s;
#pragma unroll
        for (int c = 0; c < 4; ++c) {
            float vv = fmaf(v[c], 0.9f, xv[c]);     // neuronal charge
            float sp = (vv >= 1.0f) ? 1.0f : 0.0f;  // surrogate fwd = step
            s[c] = sp;
            v[c] = (vv >= 1.0f) ? 0.0f : vv;        // hard reset to 0
        }

        // Write-once spike stream: bypass temporal caching.
        __builtin_nontemporal_store(s, y + (size_t)t * n4 + gid);
    }
}

extern "C" void kernel_launch(void* const* d_in, const int* in_sizes, int n_in,
                              void* d_out, int out_size, void* d_ws, size_t ws_size,
                              hipStream_t stream)
{
    const float* x = (const float*)d_in[0];
    float*       y = (float*)d_out;

    const unsigned total = (unsigned)in_sizes[0];     // 16*32*256*28*28
    const unsigned N     = total / T_STEPS;           // neurons per timestep
    const unsigned n4    = N / 4;                     // N divisible by 4 (28*28*256*32)

    dim3 grid((n4 + BLOCK - 1) / BLOCK), block(BLOCK);
    lif_scan_async<<<grid, block, 0, stream>>>((const v4f*)x, (v4f*)y, n4);
}